// CombinedLoss_30056181137893
// MI455X (gfx1250) — compile-verified
//
#include <hip/hip_runtime.h>

// ---------------------------------------------------------------------------
// CombinedLoss (Dice + SMSE) for [16,3,512,512] f32 -> scalar.
// Memory-bound streaming reduction (~100.6 MB read, fits MI455X 192MB L2, so
// RT-temporal loads keep replays L2-resident). Per-block 8-wave combine uses
// V_WMMA_F32_16X16X4_F32 with B=ones as a matrix-core row-sum reduction.
// Deterministic: no float atomics; block partials in d_ws, finalize sums them.
// VALU-trimmed: sigmoid via v_rcp_f32 (not IEEE div), fma-masked accumulators.
// ---------------------------------------------------------------------------

typedef __attribute__((ext_vector_type(2))) float v2f;
typedef __attribute__((ext_vector_type(8))) float v8f;

#define SMOOTH_F 0.001f
#define DICE_W   0.8f

constexpr int BB = 16, CC = 3, HH = 512, WW = 512;
constexpr int NSLAB   = BB * CC;        // 48
constexpr int HW      = HH * WW;        // 262144
constexpr int TPB     = 256;            // 8 wave32s per block
constexpr int BPS     = 32;             // blocks per slab
constexpr int N4      = HW / 4;         // 65536 float4s per slab
constexpr int STRIDE4 = BPS * TPB;      // 8192
constexpr int ITERS   = N4 / STRIDE4;   // 8
constexpr int NACC    = 7;              // sum_p,sum_t,sum_pt,cnt_pos,cnt_neg,sse_pos,sse_neg

__global__ __launch_bounds__(TPB)
void CombinedLoss_30056181137893_partial(const float* __restrict__ logits,
                                         const float* __restrict__ targets,
                                         float* __restrict__ ws)
{
    const int slab = blockIdx.x / BPS;
    const int part = blockIdx.x % BPS;

    const float4* __restrict__ lg =
        reinterpret_cast<const float4*>(logits) + (size_t)slab * N4;
    const float4* __restrict__ tg =
        reinterpret_cast<const float4*>(targets) + (size_t)slab * N4;

    float s_p = 0.f, s_t = 0.f, s_pt = 0.f;
    float c_pos = 0.f, c_neg = 0.f, e_pos = 0.f, e_neg = 0.f;

    const int base = part * TPB + (int)threadIdx.x;
#pragma unroll
    for (int it = 0; it < ITERS; ++it) {
        const int i = base + it * STRIDE4;
        const float4 x4 = lg[i];
        const float4 t4 = tg[i];
        const float xs[4] = {x4.x, x4.y, x4.z, x4.w};
        const float ts[4] = {t4.x, t4.y, t4.z, t4.w};
#pragma unroll
        for (int j = 0; j < 4; ++j) {
            const float x = xs[j], t = ts[j];
            // sigmoid via v_exp + v_rcp (2 trans ops, no IEEE div sequence)
            const float p = __builtin_amdgcn_rcpf(1.0f + __expf(-x));
            s_p += p;
            s_t += t;
            s_pt = fmaf(p, t, s_pt);
            const float dd = x - t;
            const float d2 = dd * dd;
            const float pm = (t > 0.0f)  ? 1.0f : 0.0f;   // one cndmask
            const float nm = (t == 0.0f) ? 1.0f : 0.0f;   // one cndmask
            c_pos += pm;
            c_neg += nm;
            e_pos = fmaf(pm, d2, e_pos);
            e_neg = fmaf(nm, d2, e_neg);
        }
    }

    // ---- wave32 reduction (explicit width; gfx1250 is wave32-only) ----
    float acc[NACC] = {s_p, s_t, s_pt, c_pos, c_neg, e_pos, e_neg};
#pragma unroll
    for (int off = 16; off > 0; off >>= 1) {
#pragma unroll
        for (int q = 0; q < NACC; ++q)
            acc[q] += __shfl_xor(acc[q], off, 32);
    }

    __shared__ float wpart[8 * 8];          // [wave][q], padded to 8
    const int wave = threadIdx.x >> 5;
    const int lane = threadIdx.x & 31;
    if (lane == 0) {
#pragma unroll
        for (int q = 0; q < NACC; ++q)
            wpart[wave * 8 + q] = acc[q];
    }
    __syncthreads();

    // ---- 8-wave combine via V_WMMA_F32_16X16X4_F32, B = ones ----
    // A (16x4) layout: lane L holds (M=L&15, K = (L<16 ? 0 : 2)) in VGPR0 (.x)
    // and K+1 in VGPR1 (.y). Quantity q occupies rows q (waves 0..3 over K)
    // and q+8 (waves 4..7). With B=ones: D[m][n] = sum_k A[m][k] for all n.
    const int m  = lane & 15;
    const int kb = (lane >> 4) << 1;        // 0 or 2
    v2f a; a.x = 0.0f; a.y = 0.0f;
    if (m < NACC) {
        a.x = wpart[kb * 8 + m];
        a.y = wpart[(kb + 1) * 8 + m];
    } else if (m >= 8 && m < 8 + NACC) {
        const int q = m - 8;
        a.x = wpart[(4 + kb) * 8 + q];
        a.y = wpart[(5 + kb) * 8 + q];
    }
    v2f bm; bm.x = 1.0f; bm.y = 1.0f;       // all-ones B: layout independent
    v8f cz = {};
    // executed convergently by all 8 waves (EXEC must be all 1s for WMMA)
    v8f d = __builtin_amdgcn_wmma_f32_16x16x4_f32(
        /*neg_a=*/false, a, /*neg_b=*/false, bm,
        /*c_mod=*/(short)0, cz, /*reuse_a=*/false, /*reuse_b=*/false);

    // C/D layout: VGPR q holds row q (lanes 0-15) and row q+8 (lanes 16-31);
    // total_q = row q + row q+8 -> one xor-16 shuffle.
    float tot[NACC];
#pragma unroll
    for (int q = 0; q < NACC; ++q) {
        const float v = d[q];
        tot[q] = v + __shfl_xor(v, 16, 32);
    }

    if (threadIdx.x == 0) {
        float* o = ws + (size_t)(slab * BPS + part) * 8;
#pragma unroll
        for (int q = 0; q < NACC; ++q) o[q] = tot[q];
    }
}

__global__ __launch_bounds__(64)
void CombinedLoss_30056181137893_finalize(const float* __restrict__ ws,
                                          float* __restrict__ out)
{
    __shared__ float sdice[64];
    __shared__ float smse[64];
    const int t = threadIdx.x;

    float dice = 0.0f, mse = 0.0f;
    if (t < NSLAB) {
        float a[NACC];
#pragma unroll
        for (int q = 0; q < NACC; ++q) a[q] = 0.0f;
        for (int b = 0; b < BPS; ++b) {
            const float* p = ws + (size_t)(t * BPS + b) * 8;
#pragma unroll
            for (int q = 0; q < NACC; ++q) a[q] += p[q];
        }
        const float sp = a[0], st = a[1], inter = a[2];
        const float cp = a[3], cn = a[4], ep = a[5], en = a[6];
        dice = (2.0f * inter + SMOOTH_F) / (sp + st + SMOOTH_F);
        const float mp = (cp > 0.0f) ? ep / fmaxf(cp, 1.0f) : 0.0f;
        const float mn = (cn > 0.0f) ? en / fmaxf(cn, 1.0f) : 0.0f;
        mse = mp + mn;
    }
    sdice[t] = dice;
    smse[t]  = mse;
    __syncthreads();
#pragma unroll
    for (int s = 32; s > 0; s >>= 1) {
        if (t < s) { sdice[t] += sdice[t + s]; smse[t] += smse[t + s]; }
        __syncthreads();
    }
    if (t == 0) {
        // dice_loss = sum_c(1 - mean_b dice)/C == 1 - (sum dice)/(B*C)
        const float dice_loss = 1.0f - sdice[0] / (float)NSLAB;
        const float mse_loss  = smse[0] / (float)NSLAB;   // /C/B
        out[0] = DICE_W * dice_loss + (1.0f - DICE_W) * mse_loss;
    }
}

extern "C" void kernel_launch(void* const* d_in, const int* in_sizes, int n_in,
                              void* d_out, int out_size, void* d_ws, size_t ws_size,
                              hipStream_t stream) {
    const float* logits  = (const float*)d_in[0];
    const float* targets = (const float*)d_in[1];
    float* out = (float*)d_out;
    float* ws  = (float*)d_ws;      // needs 48*32*8*4 = 48 KiB

    (void)in_sizes; (void)n_in; (void)out_size; (void)ws_size;

    CombinedLoss_30056181137893_partial<<<NSLAB * BPS, TPB, 0, stream>>>(
        logits, targets, ws);
    CombinedLoss_30056181137893_finalize<<<1, 64, 0, stream>>>(ws, out);
}